// GraphProjection_57483842289710
// MI455X (gfx1250) — compile-verified
//
#include <hip/hip_runtime.h>
#include <hip/hip_bf16.h>

typedef float v2f __attribute__((ext_vector_type(2)));
typedef float v8f __attribute__((ext_vector_type(8)));

// Bilinear interpolation of C channels from a HxHxC feature table at (xs, ys),
// executed on the matrix pipe via V_WMMA_F32_16X16X4_F32 with a broadcast-A:
//   D[m, n] = sum_k A[m,k] * B[k,n],  A[m,:] = (w11, w21, w12, w22) for all m,
//   B[:, n] = (Q11[c+n], Q21[c+n], Q12[c+n], Q22[c+n])
// f32 16x16x4 VGPR layouts (ISA 7.12.2):
//   A: vgpr0 = lanes0-15: K0, lanes16-31: K2 ; vgpr1 = K1 / K3
//   B: vgpr0 = lanes0-15: K0 row (N=lane), lanes16-31: K2 row ; vgpr1 = K1 / K3
//   D: vgpr0 = lanes0-15: M=0 (N=lane), lanes16-31: M=8 (N=lane-16)
__device__ __forceinline__ void bilinear_level_wmma(
    const float* __restrict__ feat, int H, int C,
    float xs, float ys, int lane, float* __restrict__ outp)
{
    float x1f = floorf(xs), x2f = ceilf(xs);
    float y1f = floorf(ys), y2f = ceilf(ys);
    int xi1 = min(max((int)x1f, 0), H - 1);
    int xi2 = min(max((int)x2f, 0), H - 1);
    int yi1 = min(max((int)y1f, 0), H - 1);
    int yi2 = min(max((int)y2f, 0), H - 1);

    float w11 = (x2f - xs) * (y2f - ys);
    float w21 = (xs - x1f) * (y2f - ys);
    float w12 = (x2f - xs) * (ys - y1f);
    float w22 = (xs - x1f) * (ys - y1f);

    const bool lo = (lane < 16);
    const int  n  = lane & 15;

    // A matrix: K order (w11, w21, w12, w22)
    v2f a;
    a.x = lo ? w11 : w12;   // K0 / K2
    a.y = lo ? w21 : w22;   // K1 / K3

    // B rows: K0=Q11(x1,y1) K1=Q21(x2,y1) K2=Q12(x1,y2) K3=Q22(x2,y2)
    // -> column index selected by lane half, row x1 in vgpr0, row x2 in vgpr1.
    const int col = lo ? yi1 : yi2;
    const float* r1 = feat + ((size_t)xi1 * H + col) * C;  // x1 row
    const float* r2 = feat + ((size_t)xi2 * H + col) * C;  // x2 row

    for (int cb = 0; cb < C; cb += 32) {
        v2f b0, b1;
        b0.x = r1[cb + n];        // tile 0: channels cb..cb+15
        b0.y = r2[cb + n];
        b1.x = r1[cb + 16 + n];   // tile 1: channels cb+16..cb+31
        b1.y = r2[cb + 16 + n];

        v8f c0 = {}, c1 = {};
        v8f d0 = __builtin_amdgcn_wmma_f32_16x16x4_f32(
            false, a, false, b0, (short)0, c0, false, false);
        v8f d1 = __builtin_amdgcn_wmma_f32_16x16x4_f32(
            false, a, false, b1, (short)0, c1, false, false);

        // d0[0]: lanes0-15 hold out[cb+lane]; d1[0]: lanes16-31 hold out[cb+lane]
        float val = lo ? d0[0] : d1[0];
        outp[cb + lane] = val;   // 128B coalesced store per 32 channels
    }
}

__global__ void __launch_bounds__(256)
graph_projection_kernel(const float* __restrict__ coord,
                        const float* __restrict__ f56,
                        const float* __restrict__ f28,
                        const float* __restrict__ f14,
                        const float* __restrict__ f7,
                        float* __restrict__ out, int N)
{
    const int lane      = threadIdx.x & 31;
    const int wavesPerB = blockDim.x >> 5;
    const int gwave     = blockIdx.x * wavesPerB + (threadIdx.x >> 5);
    const int nWaves    = gridDim.x * wavesPerB;

    for (int p = gwave; p < N; p += nWaves) {
        float X = coord[3 * p + 0];
        float Y = coord[3 * p + 1];
        float Z = coord[3 * p + 2];

        // h = 250*(-Y)/(-Z)+112 == 250*Y/Z+112 ; w = 250*X/(-Z)+112 == -(250*X/Z)+112
        float h = 250.0f * Y / Z + 112.0f;
        float w = -(250.0f * X / Z) + 112.0f;
        h = fminf(fmaxf(h, 0.0f), 223.0f);
        w = fminf(fmaxf(w, 0.0f), 223.0f);

        float* row = out + (size_t)p * 963;
        if (lane == 0) { row[0] = X; row[1] = Y; row[2] = Z; }

        bilinear_level_wmma(f56, 56,  64, h * 0.25f,    w * 0.25f,    lane, row + 3);
        bilinear_level_wmma(f28, 28, 128, h * 0.125f,   w * 0.125f,   lane, row + 3 + 64);
        bilinear_level_wmma(f14, 14, 256, h * 0.0625f,  w * 0.0625f,  lane, row + 3 + 192);
        bilinear_level_wmma(f7,   7, 512, h * 0.03125f, w * 0.03125f, lane, row + 3 + 448);
    }
}

extern "C" void kernel_launch(void* const* d_in, const int* in_sizes, int n_in,
                              void* d_out, int out_size, void* d_ws, size_t ws_size,
                              hipStream_t stream) {
    const float* coord = (const float*)d_in[0];
    const float* f56   = (const float*)d_in[1];
    const float* f28   = (const float*)d_in[2];
    const float* f14   = (const float*)d_in[3];
    const float* f7    = (const float*)d_in[4];
    float* out = (float*)d_out;

    const int N = in_sizes[0] / 3;   // 131072

    // One wave (32 lanes) per point, grid-stride. 256 threads = 8 waves/block.
    // 2048 blocks -> 16384 waves -> 8 points per wave; plenty of waves to
    // saturate the HBM store path (kernel is store-bandwidth bound, ~505MB out).
    const int block = 256;
    const int grid  = 2048;
    graph_projection_kernel<<<grid, block, 0, stream>>>(coord, f56, f28, f14, f7, out, N);
}